// GCN_10247791969006
// MI455X (gfx1250) — compile-verified
//
#include <hip/hip_runtime.h>

#define N_INPUT  512
#define N_HIDDEN 64
#define LDS_STRIDE 516   // 512 + 4 pad: bank = (4n + k) % 64 -> all 32 lanes x 2 words hit distinct banks

typedef __attribute__((ext_vector_type(2))) float v2f;
typedef __attribute__((ext_vector_type(4))) float v4f;
typedef __attribute__((ext_vector_type(8))) float v8f;

// ---------------------------------------------------------------------------
// Kernel 1: zero the aggregation buffer (d_out) with 128-bit stores
// ---------------------------------------------------------------------------
__global__ __launch_bounds__(256) void gcn_zero(float4* out, int n4) {
    int i = blockIdx.x * blockDim.x + threadIdx.x;
    if (i < n4) out[i] = make_float4(0.f, 0.f, 0.f, 0.f);
}

// ---------------------------------------------------------------------------
// Kernel 2: h = x * W^T + b via V_WMMA_F32_16X16X4_F32
//   - block = 256 threads = 8 waves; each wave owns a 16-node strip and
//     produces all 64 hidden channels (4 accumulators), so x is read once.
//   - W (64x512 f32 = 128 KB) staged in dynamic LDS with padded stride.
// A-frag layout (16x4 f32): lane<16 holds row M=lane, K=kb+0/kb+1 in the two
// VGPRs; lane>=16 holds row M=lane-16, K=kb+2/kb+3.  B-frag (4x16) mirrors it
// with N in the lane index.  C/D: VGPR i -> M = i + 8*(lane>=16), N = lane&15.
// ---------------------------------------------------------------------------
__global__ __launch_bounds__(256) void gcn_gemm(const float* __restrict__ x,
                                                const float* __restrict__ W,
                                                const float* __restrict__ b,
                                                float* __restrict__ h,
                                                int nNodes) {
    extern __shared__ float ldsW[];   // 64 rows * LDS_STRIDE floats

    // ---- cooperative stage of W into LDS (float4 chunks) ----
    // 64*512 floats = 8192 float4 chunks, 32 per thread
    for (int c = threadIdx.x; c < (N_HIDDEN * N_INPUT) / 4; c += 256) {
        int flat = c * 4;
        int row  = flat >> 9;          // / 512
        int col  = flat & 511;
        v4f w4 = *(const v4f*)(W + row * N_INPUT + col);
        *(v4f*)(ldsW + row * LDS_STRIDE + col) = w4;
    }
    __syncthreads();

    const int lane = threadIdx.x & 31;
    const int wv   = threadIdx.x >> 5;
    const int m0   = (blockIdx.x * 8 + wv) * 16;
    if (m0 >= nNodes) return;

    const int half = lane >> 4;        // 0: K pair {0,1}, 1: K pair {2,3}
    const int lm   = lane & 15;

    // accumulators initialized with bias: bias depends only on N = t*16 + lm
    v8f acc[4];
#pragma unroll
    for (int t = 0; t < 4; ++t) {
        float bj = b[t * 16 + lm];
        acc[t] = (v8f){bj, bj, bj, bj, bj, bj, bj, bj};
    }

    const float* arow = x + (long)(m0 + lm) * N_INPUT + half * 2;
    const float* brow = ldsW + half * 2;

#pragma unroll 4
    for (int kb = 0; kb < N_INPUT; kb += 4) {
        v2f afrag = *(const v2f*)(arow + kb);
#pragma unroll
        for (int t = 0; t < 4; ++t) {
            v2f bfrag = *(const v2f*)(brow + (t * 16 + lm) * LDS_STRIDE + kb);
            acc[t] = __builtin_amdgcn_wmma_f32_16x16x4_f32(
                false, afrag, false, bfrag, (short)0, acc[t], false, false);
        }
    }

    // write back 16x64 strip: VGPR i holds (m0 + i + 8*half, t*16 + lm)
#pragma unroll
    for (int t = 0; t < 4; ++t) {
#pragma unroll
        for (int i = 0; i < 8; ++i) {
            int m = m0 + i + half * 8;
            h[(long)m * N_HIDDEN + t * 16 + lm] = acc[t][i];
        }
    }
}

// ---------------------------------------------------------------------------
// Kernel 3: per-edge gather + scale + atomic scatter-add.
// One wave32 per edge: 2 floats per lane -> coalesced 256B read of h[col],
// 64 global_atomic_add_f32 (no return) into out[row].  h and out both live
// in the 192 MB L2, so the irregular traffic stays on-chip.
// ---------------------------------------------------------------------------
__global__ __launch_bounds__(256) void gcn_scatter(const int* __restrict__ rows,
                                                   const int* __restrict__ cols,
                                                   const float* __restrict__ vals,
                                                   const float* __restrict__ h,
                                                   float* __restrict__ out,
                                                   int nEdges) {
    int e = blockIdx.x * 8 + (threadIdx.x >> 5);
    if (e >= nEdges) return;
    int lane = threadIdx.x & 31;

    int   r = rows[e];
    int   c = cols[e];
    float v = vals[e];

    const float2 hv = *(const float2*)(h + (long)c * N_HIDDEN + lane * 2);
    float* dst = out + (long)r * N_HIDDEN + lane * 2;
    atomicAdd(dst + 0, v * hv.x);
    atomicAdd(dst + 1, v * hv.y);
}

// ---------------------------------------------------------------------------
// Kernel 4: in-place LeakyReLU(alpha) on d_out, 128-bit loads/stores
// ---------------------------------------------------------------------------
__global__ __launch_bounds__(256) void gcn_lrelu(float4* out,
                                                 const float* __restrict__ alpha,
                                                 int n4) {
    int i = blockIdx.x * blockDim.x + threadIdx.x;
    if (i >= n4) return;
    float a = alpha[0];
    float4 t = out[i];
    t.x = t.x >= 0.f ? t.x : a * t.x;
    t.y = t.y >= 0.f ? t.y : a * t.y;
    t.z = t.z >= 0.f ? t.z : a * t.z;
    t.w = t.w >= 0.f ? t.w : a * t.w;
    out[i] = t;
}

// ---------------------------------------------------------------------------
extern "C" void kernel_launch(void* const* d_in, const int* in_sizes, int n_in,
                              void* d_out, int out_size, void* d_ws, size_t ws_size,
                              hipStream_t stream) {
    const float* x     = (const float*)d_in[0];
    const int*   erow  = (const int*)d_in[1];   // jax default x64-disabled -> int32
    const int*   ecol  = (const int*)d_in[2];
    const float* evals = (const float*)d_in[3];
    const float* W     = (const float*)d_in[4];
    const float* b     = (const float*)d_in[5];
    const float* alpha = (const float*)d_in[6];

    const int nNodes = in_sizes[0] / N_INPUT;   // 50000
    const int nEdges = in_sizes[1];             // 800000
    float* out = (float*)d_out;
    float* h   = (float*)d_ws;                  // nNodes * 64 f32 = 12.8 MB scratch

    // 1) zero aggregation buffer
    int n4 = out_size / 4;
    gcn_zero<<<(n4 + 255) / 256, 256, 0, stream>>>((float4*)out, n4);

    // 2) WMMA GEMM: h = x W^T + b
    int nTiles = (nNodes + 15) / 16;            // 3125
    int gemmBlocks = (nTiles + 7) / 8;          // 391
    size_t ldsBytes = (size_t)N_HIDDEN * LDS_STRIDE * sizeof(float);  // 132096 B
    gcn_gemm<<<gemmBlocks, 256, ldsBytes, stream>>>(x, W, b, h, nNodes);

    // 3) edge gather + atomic scatter
    gcn_scatter<<<(nEdges + 7) / 8, 256, 0, stream>>>(erow, ecol, evals, h, out, nEdges);

    // 4) LeakyReLU in place
    gcn_lrelu<<<(n4 + 255) / 256, 256, 0, stream>>>((float4*)out, alpha, n4);
}